// HeteroGraphEncoder_78202764525874
// MI455X (gfx1250) — compile-verified
//
#include <hip/hip_runtime.h>

// ---------------------------------------------------------------------------
// HeteroGraphEncoder (2-layer hetero GAT) for MI455X / gfx1250, wave32.
// Dense projections use V_WMMA_F32_16X16X4_F32 (full f32 precision; workload
// is memory/atomic-bound so f32 WMMA throughput is ample).
// ---------------------------------------------------------------------------

typedef __attribute__((ext_vector_type(2))) float v2f;
typedef __attribute__((ext_vector_type(8))) float v8f;

#define N_SVC  100000
#define N_NODE 50000

static inline int cdiv_i(int a, int b) { return (a + b - 1) / b; }

// ---------------- fills ----------------
__global__ void k_fill_f32(float* __restrict__ p, float v, int n) {
    int i = blockIdx.x * blockDim.x + threadIdx.x;
    if (i < n) p[i] = v;
}
__global__ void k_fill_u32(unsigned* __restrict__ p, unsigned v, int n) {
    int i = blockIdx.x * blockDim.x + threadIdx.x;
    if (i < n) p[i] = v;
}

// ---------------- GEMM: Hout[N,M] = X[N,K] @ W[K,M], f32 WMMA ---------------
// One wave computes one 16x16 tile. A-frag layout (ISA 7.12.2, 32-bit A 16x4):
//   lanes 0-15: V0=K0, V1=K1 ; lanes 16-31: V0=K2, V1=K3 (row = lane%16)
// B (4x16) mirrored: lanes 0-15 hold K0/K1 of col=lane, lanes 16-31 K2/K3.
// C/D: VGPR j -> row M = j + 8*(lane/16), col N = lane%16.
__global__ __launch_bounds__(256)
void k_gemm_f32_wmma(const float* __restrict__ X, const float* __restrict__ W,
                     float* __restrict__ Hout, int N, int K, int M) {
    const int lane = threadIdx.x & 31;
    const int wave = threadIdx.x >> 5;
    const int kh   = lane >> 4;      // k-half (0/1)
    const int lr   = lane & 15;
    const int rowTile = blockIdx.y * 8 + wave;
    if (rowTile >= (N >> 4)) return;        // wave-uniform exit (EXEC stays full)
    const int colTile = blockIdx.x;
    const int row = (rowTile << 4) + lr;
    const int col = (colTile << 4) + lr;

    v8f acc = {};
    for (int k0 = 0; k0 < K; k0 += 4) {
        const int ka = k0 + (kh << 1);
        float2 a2 = *(const float2*)(X + (size_t)row * K + ka);
        v2f a; a[0] = a2.x; a[1] = a2.y;
        v2f b;
        b[0] = W[(size_t)ka * M + col];
        b[1] = W[(size_t)(ka + 1) * M + col];
        acc = __builtin_amdgcn_wmma_f32_16x16x4_f32(
                  false, a, false, b, (short)0, acc, false, false);
    }
#pragma unroll
    for (int j = 0; j < 8; ++j) {
        const int orow = (rowTile << 4) + j + (kh << 3);
        Hout[(size_t)orow * M + col] = acc[j];
    }
}

// ---------------- per-head attention dots: out[n,h] = sum_c H[n,h,c]*att[h,c]
__global__ void k_att_dot(const float* __restrict__ Hm, const float* __restrict__ att,
                          float* __restrict__ out, int N, int H, int C) {
    int idx = blockIdx.x * blockDim.x + threadIdx.x;
    if (idx >= N * H) return;
    int n = idx / H, h = idx - n * H;
    const float* hp = Hm + (size_t)n * H * C + (size_t)h * C;
    const float* ap = att + (size_t)h * C;
    float s = 0.f;
    for (int c = 0; c < C; c += 4) {
        float4 hv = *(const float4*)(hp + c);
        float4 av = *(const float4*)(ap + c);
        s += hv.x * av.x + hv.y * av.y + hv.z * av.z + hv.w * av.w;
    }
    out[idx] = s;
}

// monotone-orderable uint key for f32 atomic max
__device__ __forceinline__ unsigned f32key(float v) {
    unsigned u = __float_as_uint(v);
    return (u & 0x80000000u) ? ~u : (u | 0x80000000u);
}
__device__ __forceinline__ float keyf32(unsigned k) {
    unsigned u = (k & 0x80000000u) ? (k ^ 0x80000000u) : ~k;
    return __uint_as_float(u);
}
#define NEG_INF_KEY 0x007FFFFFu   // f32key(-inf)

// ---------------- edge pass 1: logits + segment max --------------------------
__global__ void k_edge_logit_max(const int* __restrict__ src, const int* __restrict__ dst,
                                 const float* __restrict__ es, const float* __restrict__ ed,
                                 float* __restrict__ e_edge, unsigned* __restrict__ mkey,
                                 int E, int H) {
    int idx = blockIdx.x * blockDim.x + threadIdx.x;
    if (idx >= E * H) return;
    int i = idx / H, h = idx - i * H;
    float v = es[src[i] * H + h] + ed[dst[i] * H + h];
    v = (v > 0.f) ? v : 0.2f * v;            // leaky_relu(0.2)
    e_edge[idx] = v;
    atomicMax(&mkey[dst[i] * H + h], f32key(v));
}

// ---------------- edge pass 2: exp + segment sum -----------------------------
__global__ void k_edge_expsum(const int* __restrict__ dst, const unsigned* __restrict__ mkey,
                              float* __restrict__ e_edge, float* __restrict__ den,
                              int E, int H) {
    int idx = blockIdx.x * blockDim.x + threadIdx.x;
    if (idx >= E * H) return;
    int i = idx / H, h = idx - i * H;
    float m = keyf32(mkey[dst[i] * H + h]);
    float ex = expf(e_edge[idx] - m);
    e_edge[idx] = ex;
    atomicAdd(&den[dst[i] * H + h], ex);
}

// ---------------- edge pass 3: normalize to alpha ----------------------------
__global__ void k_edge_alpha(const int* __restrict__ dst, const float* __restrict__ den,
                             float* __restrict__ e_edge, int E, int H) {
    int idx = blockIdx.x * blockDim.x + threadIdx.x;
    if (idx >= E * H) return;
    int i = idx / H, h = idx - i * H;
    e_edge[idx] = e_edge[idx] / (den[dst[i] * H + h] + 1e-16f);
}

// ---------------- edge pass 4: weighted scatter-add --------------------------
__global__ void k_edge_scatter(const int* __restrict__ src, const int* __restrict__ dst,
                               const float* __restrict__ alpha, const float* __restrict__ hs,
                               float* __restrict__ acc, int E, int H, int C) {
    int idx = blockIdx.x * blockDim.x + threadIdx.x;
    const int HC = H * C;
    if (idx >= E * HC) return;
    int i = idx / HC, j = idx - i * HC, h = j / C;
    float a = alpha[i * H + h];
    atomicAdd(&acc[(size_t)dst[i] * HC + j], hs[(size_t)src[i] * HC + j] * a);
}

// ---------------- relation-mean + bias + optional ELU (in place) -------------
__global__ void k_finalize(float* __restrict__ acc, const float* __restrict__ bA,
                           const float* __restrict__ bB, int N, int HC, int doElu) {
    int idx = blockIdx.x * blockDim.x + threadIdx.x;
    if (idx >= N * HC) return;
    int j = idx % HC;
    float v = 0.5f * (acc[idx] + bA[j] + bB[j]);
    if (doElu) v = (v > 0.f) ? v : expm1f(v);
    acc[idx] = v;
}

// ---------------- mean pool over rows (32 columns) ---------------------------
__global__ __launch_bounds__(256)
void k_pool_mean(const float* __restrict__ h, int N, float scale, float* __restrict__ out) {
    __shared__ float s[256];
    int col = threadIdx.x & 31;
    int rg  = threadIdx.x >> 5;   // 0..7
    float a = 0.f;
    for (int row = blockIdx.x * 8 + rg; row < N; row += gridDim.x * 8)
        a += h[(size_t)row * 32 + col];
    s[threadIdx.x] = a;
    __syncthreads();
    if (threadIdx.x < 32) {
        float v = 0.f;
        for (int g = 0; g < 8; ++g) v += s[g * 32 + col];
        atomicAdd(&out[col], v * scale);
    }
}

// ---------------------------------------------------------------------------
// Host-side orchestration
// ---------------------------------------------------------------------------
static void run_layer(const float* xs, const float* xn, int K,
                      const int* const* esrc, const int* const* edst, const int* Ecnt,
                      const float* Wsrc, const float* Wdst,
                      const float* As, const float* Ad, const float* bias,
                      int H, int C, int doElu,
                      float* acc_svc, float* acc_node,
                      float* tmp, float* es, float* ed,
                      unsigned* mkey, float* den, float* e_edge,
                      hipStream_t stream) {
    const int HC = H * C;
    const int   Nn[2]  = { N_SVC, N_NODE };
    const float* xin[2] = { xs, xn };
    float*      accp[2] = { acc_svc, acc_node };
    const int relSrc[4] = { 0, 0, 1, 1 };
    const int relDst[4] = { 0, 1, 0, 1 };

    k_fill_f32<<<cdiv_i(N_SVC * HC, 256), 256, 0, stream>>>(acc_svc, 0.f, N_SVC * HC);
    k_fill_f32<<<cdiv_i(N_NODE * HC, 256), 256, 0, stream>>>(acc_node, 0.f, N_NODE * HC);

    for (int r = 0; r < 4; ++r) {
        const int Ns = Nn[relSrc[r]], Nd = Nn[relDst[r]], E = Ecnt[r];
        const float* xsrc = xin[relSrc[r]];
        const float* xdst = xin[relDst[r]];
        float* acc = accp[relDst[r]];

        // hd -> tmp, then ed (hd itself is not needed afterwards)
        dim3 gD(HC / 16, cdiv_i(Nd / 16, 8));
        k_gemm_f32_wmma<<<gD, 256, 0, stream>>>(xdst, Wdst + (size_t)r * K * HC, tmp, Nd, K, HC);
        k_att_dot<<<cdiv_i(Nd * H, 256), 256, 0, stream>>>(tmp, Ad + (size_t)r * HC, ed, Nd, H, C);

        // hs -> tmp (overwrites hd), then es ; tmp stays live as hs for scatter
        dim3 gS(HC / 16, cdiv_i(Ns / 16, 8));
        k_gemm_f32_wmma<<<gS, 256, 0, stream>>>(xsrc, Wsrc + (size_t)r * K * HC, tmp, Ns, K, HC);
        k_att_dot<<<cdiv_i(Ns * H, 256), 256, 0, stream>>>(tmp, As + (size_t)r * HC, es, Ns, H, C);

        // segment softmax over dst
        k_fill_u32<<<cdiv_i(Nd * H, 256), 256, 0, stream>>>(mkey, NEG_INF_KEY, Nd * H);
        k_fill_f32<<<cdiv_i(Nd * H, 256), 256, 0, stream>>>(den, 0.f, Nd * H);
        k_edge_logit_max<<<cdiv_i(E * H, 256), 256, 0, stream>>>(esrc[r], edst[r], es, ed, e_edge, mkey, E, H);
        k_edge_expsum  <<<cdiv_i(E * H, 256), 256, 0, stream>>>(edst[r], mkey, e_edge, den, E, H);
        k_edge_alpha   <<<cdiv_i(E * H, 256), 256, 0, stream>>>(edst[r], den, e_edge, E, H);
        k_edge_scatter <<<cdiv_i(E * HC, 256), 256, 0, stream>>>(esrc[r], edst[r], e_edge, tmp, acc, E, H, C);
    }

    // HeteroConv mean over the two relations per dst type (+ per-relation bias)
    k_finalize<<<cdiv_i(N_SVC * HC, 256), 256, 0, stream>>>(acc_svc, bias + 0 * HC, bias + 2 * HC, N_SVC, HC, doElu);
    k_finalize<<<cdiv_i(N_NODE * HC, 256), 256, 0, stream>>>(acc_node, bias + 1 * HC, bias + 3 * HC, N_NODE, HC, doElu);
}

extern "C" void kernel_launch(void* const* d_in, const int* in_sizes, int n_in,
                              void* d_out, int out_size, void* d_ws, size_t ws_size,
                              hipStream_t stream) {
    const float* x_svc  = (const float*)d_in[0];
    const float* x_node = (const float*)d_in[1];
    const int* esrc[4] = { (const int*)d_in[2], (const int*)d_in[4], (const int*)d_in[6], (const int*)d_in[8] };
    const int* edst[4] = { (const int*)d_in[3], (const int*)d_in[5], (const int*)d_in[7], (const int*)d_in[9] };
    int Ecnt[4] = { in_sizes[2], in_sizes[4], in_sizes[6], in_sizes[8] };
    const float* W_src0   = (const float*)d_in[10];
    const float* W_dst0   = (const float*)d_in[11];
    const float* att_src0 = (const float*)d_in[12];
    const float* att_dst0 = (const float*)d_in[13];
    const float* b0       = (const float*)d_in[14];
    const float* W_src1   = (const float*)d_in[15];
    const float* W_dst1   = (const float*)d_in[16];
    const float* att_src1 = (const float*)d_in[17];
    const float* att_dst1 = (const float*)d_in[18];
    const float* b1       = (const float*)d_in[19];

    // workspace carve-up (floats); total = 41,600,000 floats = 166.4 MB
    float* ws = (float*)d_ws;
    size_t off = 0;
    auto carve = [&](size_t n) { float* p = ws + off; off += n; return p; };
    float* acc_svc0  = carve((size_t)N_SVC  * 128);
    float* acc_node0 = carve((size_t)N_NODE * 128);
    float* acc_svc1  = carve((size_t)N_SVC  * 32);
    float* acc_node1 = carve((size_t)N_NODE * 32);
    float* tmp       = carve((size_t)N_SVC  * 128);   // hd then hs, per relation
    float* es        = carve((size_t)N_SVC  * 4);
    float* ed        = carve((size_t)N_SVC  * 4);
    unsigned* mkey   = (unsigned*)carve((size_t)N_SVC * 4);
    float* den       = carve((size_t)N_SVC  * 4);
    float* e_edge    = carve((size_t)800000 * 4);

    // Layer 0: F_IN=16 -> 4 heads x 32 (concat), ELU
    run_layer(x_svc, x_node, 16, esrc, edst, Ecnt,
              W_src0, W_dst0, att_src0, att_dst0, b0,
              4, 32, /*elu*/1,
              acc_svc0, acc_node0, tmp, es, ed, mkey, den, e_edge, stream);

    // Layer 1: 128 -> 1 head x 32 (mean over 1 head == identity), no activation
    run_layer(acc_svc0, acc_node0, 128, esrc, edst, Ecnt,
              W_src1, W_dst1, att_src1, att_dst1, b1,
              1, 32, /*elu*/0,
              acc_svc1, acc_node1, tmp, es, ed, mkey, den, e_edge, stream);

    // final mean pool per node type, concat into d_out[64]
    float* out = (float*)d_out;
    k_fill_f32<<<1, 64, 0, stream>>>(out, 0.f, 64);
    k_pool_mean<<<256, 256, 0, stream>>>(acc_svc1,  N_SVC,  1.0f / (float)N_SVC,  out);
    k_pool_mean<<<256, 256, 0, stream>>>(acc_node1, N_NODE, 1.0f / (float)N_NODE, out + 32);
}